// NSScan_44667659878741
// MI455X (gfx1250) — compile-verified
//
#include <hip/hip_runtime.h>
#include <stdint.h>

// ---------------------------------------------------------------------------
// NSS (nested S-shaped scan) gather for MI455X / gfx1250.
//
// Pure permutation: out[(d*8+n), l, :] = in[n, perm_d(l) (+roll 2,2), :]
// 0 FLOPs; 805 MB out + ~1-2x 201 MB in => ~52 us at 23.3 TB/s. Bandwidth-bound.
// Tokens are whole 96-float rows (384 B, 16B-aligned): every transfer is a
// contiguous 16B chunk pushed through the CDNA5 async global<->LDS DMA path
// (ASYNCcnt). Each wave gathers 8 tokens (3072 B) with 6x
// global_load_async_to_lds_b128 (TH=NT_RT: no near reuse, keep in 192MB L2
// for the 4 direction passes), waits asynccnt, then streams them out with 6x
// global_store_async_from_lds_b128 (TH=NT: write-once output, don't pollute
// L2). s_endpgm's implicit wait-idle drains the stores.
// ---------------------------------------------------------------------------

#if defined(__has_builtin)
#  if __has_builtin(__builtin_amdgcn_global_load_async_to_lds_b128) && \
      __has_builtin(__builtin_amdgcn_global_store_async_from_lds_b128)
#    define NSS_ASYNC 1
#  endif
#endif
#ifndef NSS_ASYNC
#  define NSS_ASYNC 0
#endif

#define KH 256u
#define KW 256u
#define KC 96u
#define KN 8u
#define KL (KH * KW)            /* 65536 */
#define KTOK (4u * KN * KL)     /* 2,097,152 output tokens */
#define KTOKB (KC * 4u)         /* 384 bytes per token */

#define TOK_PER_WAVE 8u         /* 3072 B per wave = 6 b128 ops per side */
#define OPS_PER_WAVE 6u         /* (TOK_PER_WAVE*KTOKB) / (32 lanes * 16B)  */
#define WAVE_BYTES   (TOK_PER_WAVE * KTOKB)   /* 3072 */

#define CPOL_LOAD  4            /* TH = NT_RT: NT near, RT far (L2 reuse)   */
#define CPOL_STORE 1            /* TH = NT: streaming write-once output     */

// Map output token index g -> source token index (n*65536 + row*256 + col)
// g layout: g = ((d*8 + n) << 16) | l   with d in [0,4), n in [0,8), l in [0,65536)
__device__ __forceinline__ uint32_t nss_src_token(uint32_t g) {
    uint32_t l = g & 0xFFFFu;
    uint32_t n = (g >> 16) & 7u;
    uint32_t d = g >> 19;                       // 0=h_fwd 1=h_bwd 2=v_fwd 3=v_bwd
    uint32_t ll = (d & 1u) ? (0xFFFFu - l) : l; // 'bwd' reverses whole sequence
    uint32_t a  = ll >> 8;                      // major index (row for h, col for v)
    uint32_t b  = ll & 255u;                    // minor position
    uint32_t m  = (a & 1u) ? (255u - b) : b;    // (a%4)%2==1  <=>  a odd
    uint32_t row = (d < 2u) ? a : m;
    uint32_t col = (d < 2u) ? m : a;
    row = (row + 2u) & 255u;                    // roll(-2,-2)
    col = (col + 2u) & 255u;
    return (n << 16) | (row << 8) | col;
}

#if NSS_ASYNC
typedef int nss_i4 __attribute__((ext_vector_type(4)));
typedef __attribute__((address_space(1))) nss_i4 gm_i4;
typedef __attribute__((address_space(3))) nss_i4 lds_i4;

// Generic->AS3: low 32 bits of a generic LDS pointer ARE the LDS byte offset
// (aperture check truncates, ISA 10.2), so an int round-trip is exact.
__device__ __forceinline__ lds_i4* nss_as_lds(const void* p) {
    return (lds_i4*)(uint32_t)(uintptr_t)p;
}
__device__ __forceinline__ gm_i4* nss_as_global(const void* p) {
    return (gm_i4*)(uintptr_t)p;
}

// 256 threads = 8 waves; each wave moves 8 output tokens (3072 B):
//   6x global_load_async_to_lds_b128  (per-lane gathered 16B chunks)
//   s_wait_asynccnt 0
//   6x global_store_async_from_lds_b128 (contiguous 3072 B to out)
__global__ void __launch_bounds__(256)
nss_gather_async(const float* __restrict__ in, float* __restrict__ out) {
    __shared__ char smem[8 * WAVE_BYTES];     // 24 KB
    const uint32_t lane  = threadIdx.x & 31u;
    const uint32_t wave  = threadIdx.x >> 5;
    const uint32_t group = blockIdx.x * 8u + wave;     // wave's token-octet id
    const uint32_t g8    = group * TOK_PER_WAVE;       // first of 8 tokens

    const char* inB  = (const char*)in;
    char*       outB = (char*)out;
    char*       lds  = smem + wave * WAVE_BYTES;

#pragma unroll
    for (uint32_t i = 0; i < OPS_PER_WAVE; ++i) {
        uint32_t f   = i * 32u + lane;        // chunk id 0..191 within octet
        uint32_t t   = f / 24u;               // which token (0..7)
        uint32_t c16 = f - t * 24u;           // which 16B chunk of the token
        uint32_t srcByte = nss_src_token(g8 + t) * KTOKB + c16 * 16u;
        __builtin_amdgcn_global_load_async_to_lds_b128(
            nss_as_global(inB + srcByte), nss_as_lds(lds + f * 16u),
            0, CPOL_LOAD);
    }

#if defined(__has_builtin) && __has_builtin(__builtin_amdgcn_s_wait_asynccnt)
    __builtin_amdgcn_s_wait_asynccnt(0);
#else
    asm volatile("s_wait_asynccnt 0" ::: "memory");
#endif

#pragma unroll
    for (uint32_t i = 0; i < OPS_PER_WAVE; ++i) {
        uint32_t f = i * 32u + lane;
        uint32_t dstByte = g8 * KTOKB + f * 16u;   // contiguous 3072B per wave
        __builtin_amdgcn_global_store_async_from_lds_b128(
            nss_as_global(outB + dstByte), nss_as_lds(lds + f * 16u),
            0, CPOL_STORE);
    }
}
#endif // NSS_ASYNC

// Fallback: one 16B chunk per thread through VGPRs. Writes fully coalesced;
// each gathered token is still a contiguous 384B read.
__global__ void __launch_bounds__(256)
nss_gather_simple(const float4* __restrict__ in, float4* __restrict__ out) {
    uint32_t f   = blockIdx.x * 256u + threadIdx.x;  // chunk id, total KTOK*24
    uint32_t tok = f / 24u;
    uint32_t c16 = f - tok * 24u;
    uint32_t st  = nss_src_token(tok);
    float4 v = in[st * 24u + c16];
    out[f] = v;
}

extern "C" void kernel_launch(void* const* d_in, const int* in_sizes, int n_in,
                              void* d_out, int out_size, void* d_ws, size_t ws_size,
                              hipStream_t stream) {
    (void)in_sizes; (void)n_in; (void)out_size; (void)d_ws; (void)ws_size;
    const float* in = (const float*)d_in[0];
    float* out = (float*)d_out;

#if NSS_ASYNC
    // KTOK tokens / (8 tokens per wave * 8 waves per block) = 32768 blocks
    dim3 grid(KTOK / (TOK_PER_WAVE * 8u));
    nss_gather_async<<<grid, 256, 0, stream>>>(in, out);
#else
    // KTOK * 24 chunks / 256 threads = 196608 blocks
    dim3 grid((KTOK * 24u) / 256u);
    nss_gather_simple<<<grid, 256, 0, stream>>>((const float4*)in, (float4*)out);
#endif
}